// SparseVoxelConverter_73435350827171
// MI455X (gfx1250) — compile-verified
//
#include <hip/hip_runtime.h>

// SparseVoxelConverter: fused depth->xyz->voxel-index + masked compaction.
// Pure streaming kernel: 64 MB read, 128 MB written, ~8.2 us at 23.3 TB/s.
// Strategy: 4 pixels per thread, every global access is a 128-bit NT op.
// H = W = 512 are compile-time powers of two, so all index math is
// shifts/masks (no integer-division magic sequences).

typedef float v4f __attribute__((ext_vector_type(4)));

constexpr int   kH      = 512;
constexpr int   kW      = 512;
constexpr int   kHW     = kH * kW;        // 2^18
constexpr int   kLogHW  = 18;
constexpr int   kLogW   = 9;
constexpr float kCx     = 256.0f;         // W/2
constexpr float kCy     = 256.0f;         // H/2
constexpr float kInvF   = 1.0f / 256.0f;  // 1/fx == 1/fy (exact power of two)
constexpr float kMinD   = 0.1f;
constexpr float kMaxD   = 10.0f;

// truncate-toward-zero + clip [0, 63], all in float domain (0..63 exact in f32)
__device__ __forceinline__ float vox_f(float norm) {
    float t = truncf(norm);
    t = fmaxf(t, 0.0f);
    return fminf(t, 63.0f);
}

__global__ __launch_bounds__(256) void sparse_voxel_kernel(
    const float* __restrict__ rgb,     // [B,3,H,W]
    const float* __restrict__ depth,   // [B,1,H,W]
    float* __restrict__ out,           // [4N | 3N | N] floats
    int N)                             // B*H*W
{
    // Each thread owns 4 consecutive pixels (one B128-granule of each stream).
    const int g = (blockIdx.x * blockDim.x + threadIdx.x) * 4;
    if (g >= N) return;

    const int b   = g >> kLogHW;            // batch
    const int rem = g & (kHW - 1);          // pixel within image
    const int v   = rem >> kLogW;           // row
    const int u0  = rem & (kW - 1);         // first column of the 4-group

    // ---- 128-bit non-temporal loads (read-once streams) ----
    const v4f d4 = __builtin_nontemporal_load(
        reinterpret_cast<const v4f*>(depth + g));
    const int rgbBase = b * (3 * kHW) + rem;
    const v4f r4 = __builtin_nontemporal_load(
        reinterpret_cast<const v4f*>(rgb + rgbBase));
    const v4f g4 = __builtin_nontemporal_load(
        reinterpret_cast<const v4f*>(rgb + rgbBase + kHW));
    const v4f b4 = __builtin_nontemporal_load(
        reinterpret_cast<const v4f*>(rgb + rgbBase + 2 * kHW));

    float* __restrict__ out_idx = out;             // 4N floats
    float* __restrict__ out_val = out + 4 * N;     // 3N floats
    float* __restrict__ out_msk = out + 7 * N;     // N floats

    const float bf    = (float)b;
    const float urow0 = (float)u0 - kCx;
    const float yrow  = (float)v  - kCy;

    float vbuf[12];   // packed [r,g,b] x 4 points -> three B128 stores
    v4f   m4;

    #pragma unroll
    for (int j = 0; j < 4; ++j) {
        const float d = d4[j];
        const bool  m = (d > kMinD) && (d < kMaxD);

        const float ds = d * kInvF;           // exact power-of-two scale
        const float x  = (urow0 + (float)j) * ds;
        const float y  = yrow * ds;

        // norm = (coord + 1) * 32 ; trunc-toward-zero ; clip [0,63]
        const float ix = vox_f((x + 1.0f) * 32.0f);
        const float iy = vox_f((y + 1.0f) * 32.0f);
        const float iz = vox_f((d + 1.0f) * 32.0f);

        v4f idx;
        idx[0] = m ? bf : -1.0f;
        idx[1] = m ? ix : -1.0f;
        idx[2] = m ? iy : -1.0f;
        idx[3] = m ? iz : -1.0f;
        // indices row for point (g+j): 16-byte aligned B128 NT store
        __builtin_nontemporal_store(idx,
            reinterpret_cast<v4f*>(out_idx + (g + j) * 4));

        vbuf[j * 3 + 0] = m ? r4[j] : 0.0f;
        vbuf[j * 3 + 1] = m ? g4[j] : 0.0f;
        vbuf[j * 3 + 2] = m ? b4[j] : 0.0f;
        m4[j] = m ? 1.0f : 0.0f;
    }

    // values: 12 contiguous floats at out_val + g*3 (g % 4 == 0 -> 16B aligned)
    float* vp = out_val + g * 3;
    v4f w0 = { vbuf[0], vbuf[1], vbuf[2],  vbuf[3]  };
    v4f w1 = { vbuf[4], vbuf[5], vbuf[6],  vbuf[7]  };
    v4f w2 = { vbuf[8], vbuf[9], vbuf[10], vbuf[11] };
    __builtin_nontemporal_store(w0, reinterpret_cast<v4f*>(vp + 0));
    __builtin_nontemporal_store(w1, reinterpret_cast<v4f*>(vp + 4));
    __builtin_nontemporal_store(w2, reinterpret_cast<v4f*>(vp + 8));

    // mask: 4 floats, 16B aligned
    __builtin_nontemporal_store(m4, reinterpret_cast<v4f*>(out_msk + g));
}

extern "C" void kernel_launch(void* const* d_in, const int* in_sizes, int n_in,
                              void* d_out, int out_size, void* d_ws, size_t ws_size,
                              hipStream_t stream) {
    const float* rgb   = (const float*)d_in[0];   // [B,3,H,W] float32
    const float* depth = (const float*)d_in[1];   // [B,1,H,W] float32
    float* out = (float*)d_out;

    const int N = in_sizes[1];                    // depth has B*H*W elements

    const int threads = 256;                      // 8 wave32 per block
    const int groups  = (N + 3) / 4;              // 4 pixels per thread
    const int blocks  = (groups + threads - 1) / threads;

    sparse_voxel_kernel<<<blocks, threads, 0, stream>>>(rgb, depth, out, N);
}